// Mesh_Renderer_38242388804174
// MI455X (gfx1250) — compile-verified
//
#include <hip/hip_runtime.h>
#include <math.h>

#define IMG 256
#define AMBIENT 0.5f
#define DIFFUSE 0.3f
#define SPECULARC 0.2f
#define FEPS 1e-8f
#define CHUNKF 64

typedef __attribute__((ext_vector_type(2))) float v2f;
typedef __attribute__((ext_vector_type(8))) float v8f;

#if __has_builtin(__builtin_amdgcn_wmma_f32_16x16x4_f32)
#define HAVE_WMMA4 1
#else
#define HAVE_WMMA4 0
#endif

#if __has_builtin(__builtin_amdgcn_global_load_async_to_lds_b32)
#define HAVE_ASYNC 1
#else
#define HAVE_ASYNC 0
#endif

#if __has_builtin(__builtin_amdgcn_global_load_async_to_lds_b128)
#define HAVE_ASYNC128 1
#else
#define HAVE_ASYNC128 0
#endif

typedef __attribute__((address_space(1))) int gint_t;
typedef __attribute__((address_space(3))) int lint_t;
typedef int int4v __attribute__((vector_size(4 * sizeof(int))));
typedef __attribute__((address_space(1))) int4v gint4_t;
typedef __attribute__((address_space(3))) int4v lint4_t;

// Issue the async copy of one 3072B face chunk (64 faces x 48B) into LDS.
// b128 path: threads 0..191 (waves 0-5, wave-uniform) copy 16B each ->
// exactly 1 async instruction per participating wave per chunk.
__device__ __forceinline__ void issue_chunk_async(const float* chunk_src,
                                                  float* lds_dst, int t) {
#if HAVE_ASYNC128
    if (t < 192) {
        gint4_t* gp = (gint4_t*)((const char*)chunk_src + t * 16);
        lint4_t* lp = (lint4_t*)((char*)lds_dst + t * 16);
        __builtin_amdgcn_global_load_async_to_lds_b128(gp, lp, 0, 0);
    }
#elif HAVE_ASYNC
    gint_t* gp = (gint_t*)(chunk_src + t * 3);
    lint_t* lp = (lint_t*)(lds_dst + t * 3);
    __builtin_amdgcn_global_load_async_to_lds_b32(gp, lp, 0, 0);
    __builtin_amdgcn_global_load_async_to_lds_b32(gp, lp, 4, 0);
    __builtin_amdgcn_global_load_async_to_lds_b32(gp, lp, 8, 0);
#else
    lds_dst[t * 3 + 0] = chunk_src[t * 3 + 0];
    lds_dst[t * 3 + 1] = chunk_src[t * 3 + 1];
    lds_dst[t * 3 + 2] = chunk_src[t * 3 + 2];
#endif
}

// Wait until only the most recent chunk's async ops remain outstanding.
__device__ __forceinline__ void wait_async_keep_one_chunk() {
#if HAVE_ASYNC128
    __builtin_amdgcn_s_wait_asynccnt(1);
#elif HAVE_ASYNC
    __builtin_amdgcn_s_wait_asynccnt(3);
#endif
}
__device__ __forceinline__ void wait_async_all() {
#if HAVE_ASYNC128 || HAVE_ASYNC
    __builtin_amdgcn_s_wait_asynccnt(0);
#endif
}

// ---------------------------------------------------------------------------
// Kernel 1: vertex transform + projection.
// vview[v,j] = sum_i verts[v,i]*R[i,j] + T[j]  ==  [16x4 A] x [4x16 B] WMMA
// A (16x4 f32, ISA layout): V0 = K0|K2, V1 = K1|K3 split by lane half.
// B rows K hold M4[k][n] = R[k][n] (k<3) or T[n] (k==3), cols n>=3 are 0.
// ---------------------------------------------------------------------------
__global__ void k_transform(const float* __restrict__ verts,
                            const float* __restrict__ tm,
                            const float* __restrict__ fl,
                            const float* __restrict__ pp,
                            float* __restrict__ vpix,
                            int Bn, int V, int tilesPerB) {
    int b    = blockIdx.x / tilesPerB;
    int tile = blockIdx.x % tilesPerB;
    int lane = threadIdx.x;
    int m  = lane & 15;
    int hi = lane >> 4;
    int v  = tile * 16 + m;
    int vc = v < V ? v : V - 1;
    __shared__ float S[16][16];
#if HAVE_WMMA4
    {
        const float* vb = verts + (size_t)b * V * 3;
        const float* M  = tm + b * 16;
        int k0 = hi ? 2 : 0;
        v2f a, bm;
        a.x = (k0     < 3) ? vb[vc * 3 + k0]     : 1.0f;   // K in {0,2}
        a.y = (k0 + 1 < 3) ? vb[vc * 3 + k0 + 1] : 1.0f;   // K in {1,3}
        int n = m;
        float b0v = 0.0f, b1v = 0.0f;
        if (n < 3) {
            b0v = (k0     < 3) ? M[k0 * 4 + n]       : M[n * 4 + 3];
            b1v = (k0 + 1 < 3) ? M[(k0 + 1) * 4 + n] : M[n * 4 + 3];
        }
        bm.x = b0v; bm.y = b1v;
        v8f c = {0.f, 0.f, 0.f, 0.f, 0.f, 0.f, 0.f, 0.f};
        v8f d = __builtin_amdgcn_wmma_f32_16x16x4_f32(
            false, a, false, bm, (short)0, c, false, false);
#pragma unroll
        for (int j = 0; j < 8; ++j) S[j + 8 * hi][n] = d[j];
    }
    __syncthreads();
#else
    if (hi == 0) {
        const float* vb = verts + ((size_t)b * V + vc) * 3;
        const float* M  = tm + b * 16;
        for (int j = 0; j < 3; ++j)
            S[m][j] = vb[0] * M[0 * 4 + j] + vb[1] * M[1 * 4 + j] +
                      vb[2] * M[2 * 4 + j] + M[j * 4 + 3];
    }
    __syncthreads();
#endif
    if (hi == 0 && v < V) {
        float X = S[m][0], Y = S[m][1], Z = S[m][2];
        float f = fl[b], cx = pp[0], cy = pp[1];
        float* o = vpix + ((size_t)b * V + v) * 3;
        o[0] = f * X / Z + cx;
        o[1] = f * Y / Z + cy;
        o[2] = Z;
    }
}

// ---------------------------------------------------------------------------
// Kernel 2: per-face edge-function coefficients (pixel space) + world normals.
// w0 = edge(v1,v2,p) = A0*px + B0*py + C0, etc. Padded faces -> all-zero
// coefficients -> denom==0 -> never inside (matches reference zero-padding).
// ---------------------------------------------------------------------------
__global__ void k_faces(const float* __restrict__ verts,
                        const int* __restrict__ faces,
                        const float* __restrict__ vpix,
                        float* __restrict__ fpack,
                        float* __restrict__ fn,
                        int Bn, int V, int F, int Fpad) {
    int idx = blockIdx.x * blockDim.x + threadIdx.x;
    if (idx >= Bn * Fpad) return;
    int b = idx / Fpad, f = idx - b * Fpad;
    int i0 = 0, i1 = 0, i2 = 0;
    if (f < F) { i0 = faces[f * 3]; i1 = faces[f * 3 + 1]; i2 = faces[f * 3 + 2]; }
    const float* vp = vpix + (size_t)b * V * 3;
    float x0 = vp[i0 * 3], y0 = vp[i0 * 3 + 1], z0 = vp[i0 * 3 + 2];
    float x1 = vp[i1 * 3], y1 = vp[i1 * 3 + 1], z1 = vp[i1 * 3 + 2];
    float x2 = vp[i2 * 3], y2 = vp[i2 * 3 + 1], z2 = vp[i2 * 3 + 2];
    float* P = fpack + ((size_t)b * Fpad + f) * 12;
    P[0] = -(y2 - y1); P[1] = x2 - x1; P[2] = (y2 - y1) * x1 - (x2 - x1) * y1;
    P[3] = -(y0 - y2); P[4] = x0 - x2; P[5] = (y0 - y2) * x2 - (x0 - x2) * y2;
    P[6] = -(y1 - y0); P[7] = x1 - x0; P[8] = (y1 - y0) * x0 - (x1 - x0) * y0;
    P[9] = z0; P[10] = z1; P[11] = z2;
    float* N = fn + ((size_t)b * Fpad + f) * 3;
    if (f < F) {
        const float* vw = verts + (size_t)b * V * 3;
        float ax = vw[i1 * 3]     - vw[i0 * 3];
        float ay = vw[i1 * 3 + 1] - vw[i0 * 3 + 1];
        float az = vw[i1 * 3 + 2] - vw[i0 * 3 + 2];
        float bx = vw[i2 * 3]     - vw[i0 * 3];
        float by = vw[i2 * 3 + 1] - vw[i0 * 3 + 1];
        float bz = vw[i2 * 3 + 2] - vw[i0 * 3 + 2];
        N[0] = ay * bz - az * by;
        N[1] = az * bx - ax * bz;
        N[2] = ax * by - ay * bx;
    } else {
        N[0] = 0.f; N[1] = 0.f; N[2] = 0.f;
    }
}

// ---------------------------------------------------------------------------
// Kernel 3: vertex normals, deterministic per-vertex scan (no float atomics).
// Matches reference ordering: acc_k summed in face order, then (a0+a1)+a2.
// ---------------------------------------------------------------------------
__global__ void k_vnorm(const int* __restrict__ faces,
                        const float* __restrict__ fn,
                        float* __restrict__ vn,
                        int Bn, int V, int F, int Fpad) {
    int idx = blockIdx.x * blockDim.x + threadIdx.x;
    if (idx >= Bn * V) return;
    int b = idx / V, v = idx - b * V;
    const float* fb = fn + (size_t)b * Fpad * 3;
    float a0x = 0, a0y = 0, a0z = 0, a1x = 0, a1y = 0, a1z = 0, a2x = 0, a2y = 0, a2z = 0;
    for (int f = 0; f < F; ++f) {
        int j0 = faces[f * 3], j1 = faces[f * 3 + 1], j2 = faces[f * 3 + 2];
        if (j0 == v) { a0x += fb[f * 3]; a0y += fb[f * 3 + 1]; a0z += fb[f * 3 + 2]; }
        if (j1 == v) { a1x += fb[f * 3]; a1y += fb[f * 3 + 1]; a1z += fb[f * 3 + 2]; }
        if (j2 == v) { a2x += fb[f * 3]; a2y += fb[f * 3 + 1]; a2z += fb[f * 3 + 2]; }
    }
    float sx = (a0x + a1x) + a2x, sy = (a0y + a1y) + a2y, sz = (a0z + a1z) + a2z;
    float nrm = sqrtf(sx * sx + sy * sy + sz * sz) + FEPS;
    float* o = vn + ((size_t)b * V + v) * 3;
    o[0] = sx / nrm; o[1] = sy / nrm; o[2] = sz / nrm;
}

// ---------------------------------------------------------------------------
// Kernel 4: rasterize + shade. One thread per pixel; face chunks staged in
// LDS via double-buffered async global->LDS b128 loads (ASYNCcnt). Inner test
// is division-free: z-compare done by cross-multiplication with |denom|.
// ---------------------------------------------------------------------------
__global__ __launch_bounds__(256) void k_raster(
    const float* __restrict__ verts, const int* __restrict__ faces,
    const float* __restrict__ tm, const float* __restrict__ fpack,
    const float* __restrict__ vn, float* __restrict__ out,
    int Bn, int V, int F, int Fpad) {
    const int P = IMG * IMG;
    const int blocksPerB = P / 256;
    int b = blockIdx.x / blocksPerB;
    int p = (blockIdx.x % blocksPerB) * 256 + threadIdx.x;
    float px = (float)(p & (IMG - 1)) + 0.5f;
    float py = (float)(p >> 8) + 0.5f;

    __shared__ __align__(16) float fb[2][CHUNKF * 12];
    const float* src = fpack + (size_t)b * Fpad * 12;
    int nch = Fpad / CHUNKF;
    int t = threadIdx.x;

    // prefetch chunk 0
    issue_chunk_async(src, &fb[0][0], t);

    float bn = INFINITY, bd = 1.0f;  // best znum/zden (both scaled positive)
    int fid = 0, found = 0;

    for (int c = 0; c < nch; ++c) {
        int buf = c & 1;
        if (c + 1 < nch) {
            issue_chunk_async(src + (size_t)(c + 1) * CHUNKF * 12,
                              &fb[buf ^ 1][0], t);
            wait_async_keep_one_chunk();   // chunk c's async ops retired
        } else {
            wait_async_all();
        }
        __syncthreads();
        const float* fp = fb[buf];
#pragma unroll 8
        for (int i = 0; i < CHUNKF; ++i) {
            const float* q = fp + i * 12;
            float A0 = q[0], B0 = q[1], C0 = q[2];
            float A1 = q[3], B1 = q[4], C1 = q[5];
            float A2 = q[6], B2 = q[7], C2 = q[8];
            float z0 = q[9], z1 = q[10], z2 = q[11];
            float w0 = fmaf(A0, px, fmaf(B0, py, C0));
            float w1 = fmaf(A1, px, fmaf(B1, py, C1));
            float w2 = fmaf(A2, px, fmaf(B2, py, C2));
            float den = (w0 + w1) + w2;
            float ad = fabsf(den);
            float znum = fmaf(z2, w2, fmaf(z1, w1, z0 * w0));
            float snum = (den < 0.0f) ? -znum : znum;   // = ad * zint
            float mn = fminf(fminf(w0, w1), w2);
            float mx = fmaxf(fmaxf(w0, w1), w2);
            bool inside = (ad > FEPS) && ((mn >= 0.0f) || (mx <= 0.0f)) &&
                          (snum > 1e-4f * ad);
            bool win = inside && (snum * bd < bn * ad);  // zint < zbest
            if (win) { bn = snum; bd = ad; fid = c * CHUNKF + i; found = 1; }
        }
        __syncthreads();
    }

    float col = 1.0f, alpha = 0.0f;
    if (found) {
        const float* q = src + (size_t)fid * 12;
        float w0 = fmaf(q[0], px, fmaf(q[1], py, q[2]));
        float w1 = fmaf(q[3], px, fmaf(q[4], py, q[5]));
        float w2 = fmaf(q[6], px, fmaf(q[7], py, q[8]));
        float den = (w0 + w1) + w2;
        float b0 = w0 / den, b1 = w1 / den, b2 = w2 / den;
        int j0 = faces[fid * 3], j1 = faces[fid * 3 + 1], j2 = faces[fid * 3 + 2];
        const float* vw = verts + (size_t)b * V * 3;
        const float* nv = vn + (size_t)b * V * 3;
        float ppx = b0 * vw[j0 * 3]     + b1 * vw[j1 * 3]     + b2 * vw[j2 * 3];
        float ppy = b0 * vw[j0 * 3 + 1] + b1 * vw[j1 * 3 + 1] + b2 * vw[j2 * 3 + 1];
        float ppz = b0 * vw[j0 * 3 + 2] + b1 * vw[j1 * 3 + 2] + b2 * vw[j2 * 3 + 2];
        float nx = b0 * nv[j0 * 3]     + b1 * nv[j1 * 3]     + b2 * nv[j2 * 3];
        float ny = b0 * nv[j0 * 3 + 1] + b1 * nv[j1 * 3 + 1] + b2 * nv[j2 * 3 + 1];
        float nz = b0 * nv[j0 * 3 + 2] + b1 * nv[j1 * 3 + 2] + b2 * nv[j2 * 3 + 2];
        float ninv = 1.0f / (sqrtf(nx * nx + ny * ny + nz * nz) + FEPS);
        nx *= ninv; ny *= ninv; nz *= ninv;
        // cam_i = -sum_j T[j]*R[i][j]
        const float* M = tm + b * 16;
        float T0 = M[3], T1 = M[7], T2 = M[11];
        float camx = -(T0 * M[0] + T1 * M[1] + T2 * M[2]);
        float camy = -(T0 * M[4] + T1 * M[5] + T2 * M[6]);
        float camz = -(T0 * M[8] + T1 * M[9] + T2 * M[10]);
        float lx = 0.0f - ppx, ly = 0.0f - ppy, lz = 3.0f - ppz;
        float li = 1.0f / (sqrtf(lx * lx + ly * ly + lz * lz) + FEPS);
        lx *= li; ly *= li; lz *= li;
        float vx = camx - ppx, vy = camy - ppy, vz = camz - ppz;
        float vi = 1.0f / (sqrtf(vx * vx + vy * vy + vz * vz) + FEPS);
        vx *= vi; vy *= vi; vz *= vi;
        float ndl = nx * lx + ny * ly + nz * lz;
        float rx = 2.0f * ndl * nx - lx;
        float ry = 2.0f * ndl * ny - ly;
        float rz = 2.0f * ndl * nz - lz;
        float m = fmaxf(rx * vx + ry * vy + rz * vz, 0.0f);
        float m2 = m * m, m4 = m2 * m2, m8 = m4 * m4;
        float m16 = m8 * m8, m32 = m16 * m16, m64 = m32 * m32;
        float spec = (ndl > 0.0f) ? SPECULARC * m64 : 0.0f;
        col = AMBIENT + DIFFUSE * fmaxf(ndl, 0.0f) + spec;
        alpha = 1.0f;
    }
    float rgb = col * 255.0f;
    size_t PP = (size_t)P;
    float* img = out + (size_t)b * 3 * PP;
    img[0 * PP + p] = rgb;
    img[1 * PP + p] = rgb;
    img[2 * PP + p] = rgb;
    out[(size_t)Bn * 3 * PP + (size_t)b * PP + p] = alpha;
}

extern "C" void kernel_launch(void* const* d_in, const int* in_sizes, int n_in,
                              void* d_out, int out_size, void* d_ws, size_t ws_size,
                              hipStream_t stream) {
    const float* verts = (const float*)d_in[0];
    const float* tm    = (const float*)d_in[1];
    const float* fl    = (const float*)d_in[2];
    const float* pp    = (const float*)d_in[3];
    const int*   faces = (const int*)d_in[4];
    int Bn = in_sizes[1] / 16;
    int V  = in_sizes[0] / (3 * Bn);
    int F  = in_sizes[4] / 3;
    int Fpad = (F + CHUNKF - 1) / CHUNKF * CHUNKF;

    float* ws    = (float*)d_ws;
    float* vpix  = ws;                                 // B*V*3
    float* fpack = vpix + (size_t)Bn * V * 3;          // B*Fpad*12
    float* fn    = fpack + (size_t)Bn * Fpad * 12;     // B*Fpad*3
    float* vnb   = fn + (size_t)Bn * Fpad * 3;         // B*V*3

    int tiles = (V + 15) / 16;
    k_transform<<<Bn * tiles, 32, 0, stream>>>(verts, tm, fl, pp, vpix, Bn, V, tiles);

    int nf = Bn * Fpad;
    k_faces<<<(nf + 255) / 256, 256, 0, stream>>>(verts, faces, vpix, fpack, fn,
                                                  Bn, V, F, Fpad);
    int nv = Bn * V;
    k_vnorm<<<(nv + 255) / 256, 256, 0, stream>>>(faces, fn, vnb, Bn, V, F, Fpad);

    const int P = IMG * IMG;
    k_raster<<<Bn * (P / 256), 256, 0, stream>>>(verts, faces, tm, fpack, vnb,
                                                 (float*)d_out, Bn, V, F, Fpad);
}